// AttentionRGCNLayer_79156247265994
// MI455X (gfx1250) — compile-verified
//
#include <hip/hip_runtime.h>

#define N_NODES 50000
#define DIM     256
#define N_EDGES 640000
#define N_REL   200

typedef float v2f __attribute__((ext_vector_type(2)));
typedef float v8f __attribute__((ext_vector_type(8)));
typedef int   v4i __attribute__((ext_vector_type(4)));
typedef int   v8i __attribute__((ext_vector_type(8)));

__device__ __forceinline__ float dot4(float4 a, float4 b) {
    return a.x * b.x + a.y * b.y + a.z * b.z + a.w * b.w;
}

// ---- Tensor Data Mover: DMA a 2D f32 tile (tile_d1 rows x tile_d0 cols) into LDS ----
// D# group0/group1 built per CDNA5 ISA; 2D tile so
// VADDR2/VADDR3 are NULL (2-SGPR-group form). TDM ignores EXEC; issue from one wave.
__device__ __forceinline__ void tdm_load_2d(unsigned lds_off, const void* gptr,
                                            unsigned tensor_d0, unsigned tensor_d1,
                                            unsigned tile_d0, unsigned tile_d1,
                                            unsigned stride0) {
    unsigned long long ga = (unsigned long long)gptr;
    v4i g0;
    g0[0] = 1;                                            // count=1, user descriptor
    g0[1] = (int)lds_off;                                 // lds_addr (bytes)
    g0[2] = (int)(unsigned)(ga & 0xFFFFFFFFull);          // global_addr[31:0]
    g0[3] = (int)((unsigned)((ga >> 32) & 0x1FFFFFFull) | (2u << 30)); // ga[56:32] | type=2
    v8i g1;
    g1[0] = (int)(2u << 16);                              // data_size=4B, no multicast
    g1[1] = (int)((tensor_d0 & 0xFFFFu) << 16);           // tensor_dim0[15:0]
    g1[2] = (int)(((tensor_d0 >> 16) & 0xFFFFu) | ((tensor_d1 & 0xFFFFu) << 16));
    g1[3] = (int)(((tensor_d1 >> 16) & 0xFFFFu) | ((tile_d0 & 0xFFFFu) << 16));
    g1[4] = (int)(tile_d1 & 0xFFFFu);                     // tile_dim1, tile_dim2=0
    g1[5] = (int)stride0;                                 // tensor_dim0_stride[31:0]
    g1[6] = 0;
    g1[7] = 0;
    asm volatile("tensor_load_to_lds %0, %1" :: "s"(g0), "s"(g1) : "memory");
}

// ---------------- W = (W3 @ a)[:,0]  (768 outputs, tiny) ----------------
__global__ void k_w3a(const float* __restrict__ W3, const float* __restrict__ a,
                      float* __restrict__ Wv) {
    int i = blockIdx.x * blockDim.x + threadIdx.x;
    if (i >= 3 * DIM) return;
    const float* row = W3 + (size_t)i * (3 * DIM);
    float s = 0.f;
    for (int j = 0; j < 3 * DIM; ++j) s += row[j] * a[j];
    Wv[i] = s;
}

// ------- per-node s_src/s_dst and per-relation s_rel dot products -------
__global__ void k_scalars(const float* __restrict__ h, const float* __restrict__ rel_emb,
                          const float* __restrict__ Wv,
                          float* __restrict__ s_src, float* __restrict__ s_dst,
                          float* __restrict__ s_rel) {
    int row  = (blockIdx.x * blockDim.x + threadIdx.x) >> 5;
    int lane = threadIdx.x & 31;
    if (row >= N_NODES + N_REL) return;
    const int base = lane * 8;
    if (row < N_NODES) {
        float4 h0 = *(const float4*)(h + (size_t)row * DIM + base);
        float4 h1 = *(const float4*)(h + (size_t)row * DIM + base + 4);
        float4 ws0 = *(const float4*)(Wv + base);
        float4 ws1 = *(const float4*)(Wv + base + 4);
        float4 wd0 = *(const float4*)(Wv + DIM + base);
        float4 wd1 = *(const float4*)(Wv + DIM + base + 4);
        float ss = dot4(h0, ws0) + dot4(h1, ws1);
        float sd = dot4(h0, wd0) + dot4(h1, wd1);
        for (int o = 16; o > 0; o >>= 1) {
            ss += __shfl_xor(ss, o, 32);
            sd += __shfl_xor(sd, o, 32);
        }
        if (lane == 0) { s_src[row] = ss; s_dst[row] = sd; }
    } else {
        int r = row - N_NODES;
        float4 r0 = *(const float4*)(rel_emb + (size_t)r * DIM + base);
        float4 r1 = *(const float4*)(rel_emb + (size_t)r * DIM + base + 4);
        float4 wr0 = *(const float4*)(Wv + 2 * DIM + base);
        float4 wr1 = *(const float4*)(Wv + 2 * DIM + base + 4);
        float sr = dot4(r0, wr0) + dot4(r1, wr1);
        for (int o = 16; o > 0; o >>= 1) sr += __shfl_xor(sr, o, 32);
        if (lane == 0) s_rel[r] = sr;
    }
}

// ---- edge scatter: acc[dst] += alpha*(h[src]+rel_emb[etype]); deg[dst]++ ----
__global__ void k_edges(const float* __restrict__ h, const float* __restrict__ rel_emb,
                        const int* __restrict__ src, const int* __restrict__ dst,
                        const int* __restrict__ etype,
                        const float* __restrict__ s_src, const float* __restrict__ s_dst,
                        const float* __restrict__ s_rel,
                        float* __restrict__ acc, float* __restrict__ deg) {
    int e    = (blockIdx.x * blockDim.x + threadIdx.x) >> 5;
    int lane = threadIdx.x & 31;
    if (e >= N_EDGES) return;
    int s = src[e], d = dst[e], r = etype[e];
    float score = s_src[s] + s_dst[d] + s_rel[r];
    float lr    = score >= 0.f ? score : 0.01f * score;   // jax leaky_relu slope 0.01
    float ex    = __expf(lr);
    float alpha = ex / ex;                                 // softmax over length-1 axis
    const int base = lane * 8;
    float4 h0 = *(const float4*)(h + (size_t)s * DIM + base);
    float4 h1 = *(const float4*)(h + (size_t)s * DIM + base + 4);
    float4 r0 = *(const float4*)(rel_emb + (size_t)r * DIM + base);
    float4 r1 = *(const float4*)(rel_emb + (size_t)r * DIM + base + 4);
    float* ap = acc + (size_t)d * DIM + base;
    atomicAdd(ap + 0, (h0.x + r0.x) * alpha);
    atomicAdd(ap + 1, (h0.y + r0.y) * alpha);
    atomicAdd(ap + 2, (h0.z + r0.z) * alpha);
    atomicAdd(ap + 3, (h0.w + r0.w) * alpha);
    atomicAdd(ap + 4, (h1.x + r1.x) * alpha);
    atomicAdd(ap + 5, (h1.y + r1.y) * alpha);
    atomicAdd(ap + 6, (h1.z + r1.z) * alpha);
    atomicAdd(ap + 7, (h1.w + r1.w) * alpha);
    if (lane == 0) atomicAdd(deg + d, 1.0f);
}

// ---- fused output: out = relu(acc@Wn) + h @ (deg>0 ? Wl : We) ----
// 32-row band per block; TDM stages 64-K slabs of the three weights (+A tiles)
// into LDS; 8 waves x 2 col tiles x 2 row tiles of f32 WMMA 16x16x4.
#define OFF_WN  0u
#define OFF_WL  65536u
#define OFF_WE  131072u
#define OFF_ACC 196608u
#define OFF_H   204800u
#define LDS_BYTES 212992u

__global__ void __launch_bounds__(256) k_out(const float* __restrict__ h,
                                             const float* __restrict__ Wn,
                                             const float* __restrict__ Wl,
                                             const float* __restrict__ We,
                                             const float* __restrict__ acc,
                                             const float* __restrict__ deg,
                                             float* __restrict__ out) {
    extern __shared__ float smem[];
    float* s_wn  = smem;            // 64 x 256
    float* s_wl  = smem + 16384;    // 64 x 256
    float* s_we  = smem + 32768;    // 64 x 256
    float* s_acc = smem + 49152;    // 32 x 64
    float* s_h   = smem + 51200;    // 32 x 64

    const int lane = threadIdx.x & 31;
    const int wave = threadIdx.x >> 5;
    const int row0 = blockIdx.x * 32;
    const int rowi = lane & 15;
    const int koff = (lane >> 4) << 1;          // f32 WMMA A frag: K = k+{0,1}/{2,3}

    v8f Cn[2][2] = {}; v8f Cl[2][2] = {}; v8f Ce[2][2] = {};

    for (int stg = 0; stg < 4; ++stg) {
        const int kc = stg * 64;
        if (wave == 0) {
            tdm_load_2d(OFF_WN,  Wn + (size_t)kc * DIM, DIM, DIM, DIM, 64, DIM);
            tdm_load_2d(OFF_WL,  Wl + (size_t)kc * DIM, DIM, DIM, DIM, 64, DIM);
            tdm_load_2d(OFF_WE,  We + (size_t)kc * DIM, DIM, DIM, DIM, 64, DIM);
            tdm_load_2d(OFF_ACC, acc + (size_t)row0 * DIM + kc, DIM, N_NODES, 64, 32, DIM);
            tdm_load_2d(OFF_H,   h   + (size_t)row0 * DIM + kc, DIM, N_NODES, 64, 32, DIM);
            __builtin_amdgcn_s_wait_tensorcnt(0);
        }
        __syncthreads();
        #pragma unroll 4
        for (int k = 0; k < 64; k += 4) {
            const int kk = k + koff;
            v2f aA[2], aH[2];
            #pragma unroll
            for (int rt = 0; rt < 2; ++rt) {
                aA[rt] = *(const v2f*)(s_acc + (rt * 16 + rowi) * 64 + kk);
                aH[rt] = *(const v2f*)(s_h   + (rt * 16 + rowi) * 64 + kk);
            }
            #pragma unroll
            for (int ct = 0; ct < 2; ++ct) {
                const int col = (wave * 2 + ct) * 16 + rowi;
                v2f bN, bL, bE;
                bN.x = s_wn[kk * DIM + col]; bN.y = s_wn[(kk + 1) * DIM + col];
                bL.x = s_wl[kk * DIM + col]; bL.y = s_wl[(kk + 1) * DIM + col];
                bE.x = s_we[kk * DIM + col]; bE.y = s_we[(kk + 1) * DIM + col];
                #pragma unroll
                for (int rt = 0; rt < 2; ++rt) {
                    Cn[ct][rt] = __builtin_amdgcn_wmma_f32_16x16x4_f32(false, aA[rt], false, bN, (short)0, Cn[ct][rt], false, false);
                    Cl[ct][rt] = __builtin_amdgcn_wmma_f32_16x16x4_f32(false, aH[rt], false, bL, (short)0, Cl[ct][rt], false, false);
                    Ce[ct][rt] = __builtin_amdgcn_wmma_f32_16x16x4_f32(false, aH[rt], false, bE, (short)0, Ce[ct][rt], false, false);
                }
            }
        }
        __syncthreads();
    }

    const int mhalf = (lane >> 4) << 3;         // lanes 16-31 hold rows M+8
    #pragma unroll
    for (int ct = 0; ct < 2; ++ct) {
        const int col = (wave * 2 + ct) * 16 + rowi;
        #pragma unroll
        for (int rt = 0; rt < 2; ++rt) {
            #pragma unroll
            for (int v = 0; v < 8; ++v) {
                const int m = row0 + rt * 16 + mhalf + v;
                if (m < N_NODES) {
                    float cnv = Cn[ct][rt][v];
                    float agg = cnv > 0.f ? cnv : 0.f;
                    float lp  = deg[m] > 0.f ? Cl[ct][rt][v] : Ce[ct][rt][v];
                    out[(size_t)m * DIM + col] = agg + lp;
                }
            }
        }
    }
}

extern "C" void kernel_launch(void* const* d_in, const int* in_sizes, int n_in,
                              void* d_out, int out_size, void* d_ws, size_t ws_size,
                              hipStream_t stream) {
    const float* h       = (const float*)d_in[0];
    const float* rel_emb = (const float*)d_in[1];
    const float* Wn      = (const float*)d_in[2];
    const float* a       = (const float*)d_in[3];
    const float* W3      = (const float*)d_in[4];
    const float* Wl      = (const float*)d_in[5];
    const float* We      = (const float*)d_in[6];
    const int*   src     = (const int*)d_in[7];
    const int*   dst     = (const int*)d_in[8];
    const int*   etype   = (const int*)d_in[9];
    float* out = (float*)d_out;

    float* ws    = (float*)d_ws;
    float* acc   = ws;                                 // N*D
    float* deg   = acc + (size_t)N_NODES * DIM;        // N
    float* s_src = deg + N_NODES;                      // N
    float* s_dst = s_src + N_NODES;                    // N
    float* s_rel = s_dst + N_NODES;                    // R
    float* Wv    = s_rel + N_REL;                      // 768

    hipMemsetAsync(acc, 0, (size_t)(N_NODES * DIM + N_NODES) * sizeof(float), stream);

    k_w3a<<<(3 * DIM + 255) / 256, 256, 0, stream>>>(W3, a, Wv);

    int rows = N_NODES + N_REL;
    k_scalars<<<(rows + 7) / 8, 256, 0, stream>>>(h, rel_emb, Wv, s_src, s_dst, s_rel);

    k_edges<<<(N_EDGES + 7) / 8, 256, 0, stream>>>(h, rel_emb, src, dst, etype,
                                                   s_src, s_dst, s_rel, acc, deg);

    k_out<<<(N_NODES + 31) / 32, 256, LDS_BYTES, stream>>>(h, Wn, Wl, We, acc, deg, out);
}